// Seq2SeqAttention_45320494907498
// MI455X (gfx1250) — compile-verified
//
#include <hip/hip_runtime.h>
#include <math.h>

// Problem constants (from reference)
#define BDIM 64
#define SDIM 400
#define TDIM 25
#define EDIM 256
#define HDIM 512
#define H2DIM 256
#define MDIM 100
#define VDIM 50000
#define EOS_TOK 2

typedef __attribute__((ext_vector_type(2))) float v2f;
typedef __attribute__((ext_vector_type(8))) float v8f;

__device__ __forceinline__ float sigmoidf_(float x) { return 1.0f / (1.0f + expf(-x)); }

// ---------------------------------------------------------------------------
// Tiled fp32 WMMA GEMM:  C[M,N] = A[M,K] @ W[N,K]^T (+ bias[N])
// Block: 256 threads (8 wave32 waves), 64x64 C tile, K stepped 16 via LDS,
// inner K stepped 4 via v_wmma_f32_16x16x4_f32. Each wave computes two
// adjacent 16x16 tiles sharing one A fragment. Interior blocks use
// b128 global loads + b128 LDS stores; edge blocks use guarded scalars.
// ---------------------------------------------------------------------------
__global__ __launch_bounds__(256)
void gemm_wmma_f32(const float* __restrict__ A, int lda,
                   const float* __restrict__ W,   // (N,K) row-major
                   const float* __restrict__ bias, // may be null
                   float* __restrict__ C, int ldc,
                   int Mr, int Nc, int K)
{
    // pad 20: float4 LDS stores stay 16B aligned; 20*m mod 64 distinct for
    // m=0..15 -> conflict-free fragment reads
    __shared__ float As[64][20];
    __shared__ float Ws[64][20];

    const int tid  = threadIdx.x;
    const int lane = tid & 31;
    const int wave = tid >> 5;
    const int rowBase = blockIdx.y * 64;
    const int colBase = blockIdx.x * 64;

    const int tm  = wave >> 1;        // 0..3  (16-row tile within block)
    const int tn0 = (wave & 1) * 2;   // 0 or 2 (pair of 16-col tiles)
    const int hh  = lane >> 4;        // half-wave select
    const int mn  = lane & 15;        // row/col within 16x16 fragment

    const bool interior = (rowBase + 64 <= Mr) && (colBase + 64 <= Nc);
    const int lr  = tid >> 2;         // 0..63 (row for vector load)
    const int lc4 = (tid & 3) * 4;    // 0,4,8,12

    v8f acc0 = {};
    v8f acc1 = {};

    for (int kb = 0; kb < K; kb += 16) {
        if (interior) {
            // 256 threads x one float4 each per tile: global b128 -> LDS b128
            const float4 av = *(const float4*)(A + (size_t)(rowBase + lr) * lda + kb + lc4);
            const float4 wv = *(const float4*)(W + (size_t)(colBase + lr) * K + kb + lc4);
            *(float4*)&As[lr][lc4] = av;
            *(float4*)&Ws[lr][lc4] = wv;
        } else {
            #pragma unroll
            for (int i = 0; i < 4; ++i) {
                int idx = i * 256 + tid;
                int r = idx >> 4, cc = idx & 15;
                int gr = rowBase + r;
                As[r][cc] = (gr < Mr) ? A[(size_t)gr * lda + kb + cc] : 0.0f;
                int gn = colBase + r;
                Ws[r][cc] = (gn < Nc) ? W[(size_t)gn * K + kb + cc] : 0.0f;
            }
        }
        __syncthreads();

        #pragma unroll
        for (int ks = 0; ks < 16; ks += 4) {
            int kk = ks + 2 * hh;
            v2f a, b0, b1;
            // A 16x4 fragment: lane m = mn, VGPR0/1 hold K = 2*hh, 2*hh+1
            a.x  = As[tm * 16 + mn][kk];
            a.y  = As[tm * 16 + mn][kk + 1];
            // B 4x16 fragment: B[k][n] = W[n][k]
            b0.x = Ws[tn0 * 16 + mn][kk];
            b0.y = Ws[tn0 * 16 + mn][kk + 1];
            b1.x = Ws[(tn0 + 1) * 16 + mn][kk];
            b1.y = Ws[(tn0 + 1) * 16 + mn][kk + 1];
            acc0 = __builtin_amdgcn_wmma_f32_16x16x4_f32(false, a, false, b0,
                                                         (short)0, acc0, false, false);
            acc1 = __builtin_amdgcn_wmma_f32_16x16x4_f32(false, a, false, b1,
                                                         (short)0, acc1, false, false);
        }
        __syncthreads();
    }

    // Store: C VGPR i -> row (i + 8*hh), col mn within the 16x16 tile
    #pragma unroll
    for (int i = 0; i < 8; ++i) {
        int gr  = rowBase + tm * 16 + i + 8 * hh;
        int gc0 = colBase + tn0 * 16 + mn;
        int gc1 = colBase + (tn0 + 1) * 16 + mn;
        if (gr < Mr) {
            if (gc0 < Nc) C[(size_t)gr * ldc + gc0] = acc0[i] + (bias ? bias[gc0] : 0.0f);
            if (gc1 < Nc) C[(size_t)gr * ldc + gc1] = acc1[i] + (bias ? bias[gc1] : 0.0f);
        }
    }
}

// ---------------------------------------------------------------------------
// Elementwise / small kernels
// ---------------------------------------------------------------------------
__global__ void zero_f(float* p, int n)
{
    int i = blockIdx.x * blockDim.x + threadIdx.x;
    if (i < n) p[i] = 0.0f;
}

// X[s][b][e] = emb[src[b][s]][e]   (time-major embeddings)
__global__ void embed_src(const int* __restrict__ src, const float* __restrict__ emb,
                          float* __restrict__ X, int total)
{
    int idx = blockIdx.x * blockDim.x + threadIdx.x;
    if (idx >= total) return;
    int e = idx % EDIM;
    int r = idx / EDIM;       // r = s*B + b
    int b = r % BDIM;
    int s = r / BDIM;
    X[idx] = emb[(size_t)src[b * SDIM + s] * EDIM + e];
}

// One GRU cell update (gi/gh already include their biases).
// h updated in place; optionally writes hidden into ys[b*HDIM + col0 + j].
__global__ void gru_update(const float* __restrict__ gi, const float* __restrict__ gh,
                           float* __restrict__ h, float* __restrict__ ys,
                           int Hd, int col0)
{
    int idx = blockIdx.x * blockDim.x + threadIdx.x;
    if (idx >= BDIM * Hd) return;
    int b = idx / Hd, j = idx % Hd;
    const float* gib = gi + (size_t)b * 3 * Hd;
    const float* ghb = gh + (size_t)b * 3 * Hd;
    float r = sigmoidf_(gib[j] + ghb[j]);
    float z = sigmoidf_(gib[Hd + j] + ghb[Hd + j]);
    float n = tanhf(gib[2 * Hd + j] + r * ghb[2 * Hd + j]);
    float h2 = (1.0f - z) * n + z * h[idx];
    h[idx] = h2;
    if (ys) ys[(size_t)b * HDIM + col0 + j] = h2;
}

// enc_hidden = concat(hf, hb); also seeds decoder hidden
__global__ void concat_h(const float* __restrict__ hf, const float* __restrict__ hb,
                         float* __restrict__ ench, float* __restrict__ hdec)
{
    int idx = blockIdx.x * blockDim.x + threadIdx.x;
    if (idx >= BDIM * HDIM) return;
    int b = idx / HDIM, j = idx % HDIM;
    float v = (j < H2DIM) ? hf[b * H2DIM + j] : hb[b * H2DIM + (j - H2DIM)];
    ench[idx] = v;
    hdec[idx] = v;
}

// states *= sigmoid(SG + G2[b])   (SG includes b1, G2 includes b2)
__global__ void apply_sgate(float* __restrict__ states, const float* __restrict__ SG,
                            const float* __restrict__ G2, int total)
{
    int idx = blockIdx.x * blockDim.x + threadIdx.x;
    if (idx >= total) return;
    int j = idx % HDIM;
    int b = (idx / HDIM) % BDIM;
    states[idx] *= sigmoidf_(SG[idx] + G2[b * HDIM + j]);
}

// e[s][b] = sum_m tanh(dh[b][m] + eproj[s][b][m]) * Wo[m] + bo  (one wave per (s,b))
__global__ void attn_scores(const float* __restrict__ dh, const float* __restrict__ eproj,
                            const float* __restrict__ Wo, const float* __restrict__ bo,
                            float* __restrict__ e)
{
    int wid  = (blockIdx.x * blockDim.x + threadIdx.x) >> 5;
    int lane = threadIdx.x & 31;
    if (wid >= SDIM * BDIM) return;
    int s = wid / BDIM, b = wid % BDIM;
    const float* ep  = eproj + (size_t)(s * BDIM + b) * MDIM;
    const float* dhb = dh + (size_t)b * MDIM;
    float p = 0.0f;
    for (int m = lane; m < MDIM; m += 32)
        p += tanhf(dhb[m] + ep[m]) * Wo[m];
    for (int off = 16; off > 0; off >>= 1)
        p += __shfl_down(p, off, 32);
    if (lane == 0) e[s * BDIM + b] = p + bo[0];
}

// Per-batch softmax over S and context c[b] = sum_s a[s,b] * states[s,b,:]
__global__ __launch_bounds__(256)
void softmax_ctx(const float* __restrict__ e, const float* __restrict__ states,
                 float* __restrict__ c)
{
    __shared__ float red[256];
    __shared__ float sa[SDIM];
    int b = blockIdx.x, tid = threadIdx.x;

    float m = -1e30f;
    for (int s = tid; s < SDIM; s += 256) m = fmaxf(m, e[s * BDIM + b]);
    red[tid] = m; __syncthreads();
    for (int o = 128; o > 0; o >>= 1) {
        if (tid < o) red[tid] = fmaxf(red[tid], red[tid + o]);
        __syncthreads();
    }
    m = red[0]; __syncthreads();

    float sum = 0.0f;
    for (int s = tid; s < SDIM; s += 256) {
        float ex = expf(e[s * BDIM + b] - m);
        sa[s] = ex;
        sum += ex;
    }
    red[tid] = sum; __syncthreads();
    for (int o = 128; o > 0; o >>= 1) {
        if (tid < o) red[tid] += red[tid + o];
        __syncthreads();
    }
    float inv = 1.0f / red[0];

    for (int j = tid; j < HDIM; j += 256) {
        float acc = 0.0f;
        for (int s = 0; s < SDIM; ++s)
            acc += sa[s] * states[(size_t)(s * BDIM + b) * HDIM + j];
        c[b * HDIM + j] = acc * inv;
    }
}

// xcat[b] = concat(c[b], emb[trg[b][t]])
__global__ void build_xcat(const float* __restrict__ c, const int* __restrict__ trg,
                           const float* __restrict__ emb, float* __restrict__ xcat, int t)
{
    int idx = blockIdx.x * blockDim.x + threadIdx.x;
    if (idx >= BDIM * (HDIM + EDIM)) return;
    int b = idx / (HDIM + EDIM), j = idx % (HDIM + EDIM);
    xcat[idx] = (j < HDIM)
        ? c[b * HDIM + j]
        : emb[(size_t)trg[b * TDIM + t] * EDIM + (j - HDIM)];
}

// Per-row log-softmax NLL over the vocabulary
__global__ __launch_bounds__(256)
void loss_row(const float* __restrict__ logits, const int* __restrict__ trg,
              int t, float* __restrict__ nll, float* __restrict__ wb)
{
    __shared__ float red[256];
    int b = blockIdx.x, tid = threadIdx.x;
    const float* row = logits + (size_t)b * VDIM;

    float m = -1e30f;
    for (int v = tid; v < VDIM; v += 256) m = fmaxf(m, row[v]);
    red[tid] = m; __syncthreads();
    for (int o = 128; o > 0; o >>= 1) {
        if (tid < o) red[tid] = fmaxf(red[tid], red[tid + o]);
        __syncthreads();
    }
    m = red[0]; __syncthreads();

    float sum = 0.0f;
    for (int v = tid; v < VDIM; v += 256) sum += expf(row[v] - m);
    red[tid] = sum; __syncthreads();
    for (int o = 128; o > 0; o >>= 1) {
        if (tid < o) red[tid] += red[tid + o];
        __syncthreads();
    }

    if (tid == 0) {
        int tok = trg[b * TDIM + t];
        nll[b] = (m + logf(red[0])) - row[tok];
        wb[b]  = (tok != EOS_TOK) ? 1.0f : 0.0f;
    }
}

// acc += sum(w*nll) / sum(w)
__global__ void loss_accum(const float* __restrict__ nll, const float* __restrict__ wb,
                           float* __restrict__ acc)
{
    __shared__ float s1[64];
    __shared__ float s2[64];
    int tid = threadIdx.x;  // 64 threads
    s1[tid] = wb[tid] * nll[tid];
    s2[tid] = wb[tid];
    __syncthreads();
    for (int o = 32; o > 0; o >>= 1) {
        if (tid < o) { s1[tid] += s1[tid + o]; s2[tid] += s2[tid + o]; }
        __syncthreads();
    }
    if (tid == 0) acc[0] += s1[0] / s2[0];
}

__global__ void finalize_loss(const float* __restrict__ acc, float* __restrict__ out)
{
    if (threadIdx.x == 0 && blockIdx.x == 0) out[0] = acc[0] / (float)BDIM;
}

// ---------------------------------------------------------------------------
// Host orchestration
// ---------------------------------------------------------------------------
extern "C" void kernel_launch(void* const* d_in, const int* in_sizes, int n_in,
                              void* d_out, int out_size, void* d_ws, size_t ws_size,
                              hipStream_t stream)
{
    const int*   src   = (const int*)  d_in[0];
    const int*   trg   = (const int*)  d_in[1];
    const float* emb   = (const float*)d_in[2];
    const float* Wih_f = (const float*)d_in[3];
    const float* Whh_f = (const float*)d_in[4];
    const float* bih_f = (const float*)d_in[5];
    const float* bhh_f = (const float*)d_in[6];
    const float* Wih_b = (const float*)d_in[7];
    const float* Whh_b = (const float*)d_in[8];
    const float* bih_b = (const float*)d_in[9];
    const float* bhh_b = (const float*)d_in[10];
    const float* W1    = (const float*)d_in[11];
    const float* b1    = (const float*)d_in[12];
    const float* W2    = (const float*)d_in[13];
    const float* b2    = (const float*)d_in[14];
    const float* Wd    = (const float*)d_in[15];
    const float* bd    = (const float*)d_in[16];
    const float* We    = (const float*)d_in[17];
    const float* be    = (const float*)d_in[18];
    const float* Wo    = (const float*)d_in[19];
    const float* bo    = (const float*)d_in[20];
    const float* Wih_c = (const float*)d_in[21];
    const float* Whh_c = (const float*)d_in[22];
    const float* bih_c = (const float*)d_in[23];
    const float* bhh_c = (const float*)d_in[24];
    const float* Wv    = (const float*)d_in[25];
    const float* bv    = (const float*)d_in[26];
    float* out = (float*)d_out;

    // Workspace layout (fp32 elements, 256B aligned chunks)
    float* ws = (float*)d_ws;
    size_t off = 0;
    auto alloc = [&](size_t n) { float* p = ws + off; off += (n + 63) & ~(size_t)63; return p; };

    float* X      = alloc((size_t)SDIM * BDIM * EDIM);       // 26.2 MB (reused as LOGITS)
    float* GI     = alloc((size_t)SDIM * BDIM * 3 * H2DIM);  // 78.6 MB (reused as SG)
    float* STATES = alloc((size_t)SDIM * BDIM * HDIM);       // 52.4 MB
    float* EPROJ  = alloc((size_t)SDIM * BDIM * MDIM);       // 10.2 MB
    float* G2     = alloc((size_t)BDIM * HDIM);
    float* HF     = alloc((size_t)BDIM * H2DIM);
    float* HB     = alloc((size_t)BDIM * H2DIM);
    float* ENCH   = alloc((size_t)BDIM * HDIM);
    float* HDEC   = alloc((size_t)BDIM * HDIM);
    float* GH     = alloc((size_t)BDIM * 3 * HDIM);          // shared gh scratch
    float* DH     = alloc((size_t)BDIM * MDIM);
    float* SCR    = alloc((size_t)SDIM * BDIM);
    float* CVEC   = alloc((size_t)BDIM * HDIM);
    float* XCAT   = alloc((size_t)BDIM * (HDIM + EDIM));
    float* GIC    = alloc((size_t)BDIM * 3 * HDIM);
    float* GHC    = alloc((size_t)BDIM * 3 * HDIM);
    float* NLL    = alloc((size_t)BDIM);
    float* WB     = alloc((size_t)BDIM);
    float* LACC   = alloc((size_t)64);
    float* SG     = GI;   // reuse: GI dead after encoder scans
    float* LOGITS = X;    // reuse: X dead after GI_b GEMM

    auto gemm = [&](const float* A, int lda, const float* W, const float* bias,
                    float* C, int ldc, int Mr, int Nc, int K) {
        dim3 g((Nc + 63) / 64, (Mr + 63) / 64);
        gemm_wmma_f32<<<g, dim3(256), 0, stream>>>(A, lda, W, bias, C, ldc, Mr, Nc, K);
    };

    // ---- init ----
    zero_f<<<(BDIM * H2DIM + 255) / 256, 256, 0, stream>>>(HF, BDIM * H2DIM);
    zero_f<<<(BDIM * H2DIM + 255) / 256, 256, 0, stream>>>(HB, BDIM * H2DIM);
    zero_f<<<1, 64, 0, stream>>>(LACC, 1);

    // ---- embeddings (time-major) ----
    {
        int total = SDIM * BDIM * EDIM;
        embed_src<<<(total + 255) / 256, 256, 0, stream>>>(src, emb, X, total);
    }

    // ---- forward encoder: precomputed input GEMM, then sequential scan ----
    gemm(X, EDIM, Wih_f, bih_f, GI, 3 * H2DIM, SDIM * BDIM, 3 * H2DIM, EDIM);
    for (int t = 0; t < SDIM; ++t) {
        gemm(HF, H2DIM, Whh_f, bhh_f, GH, 3 * H2DIM, BDIM, 3 * H2DIM, H2DIM);
        gru_update<<<(BDIM * H2DIM + 255) / 256, 256, 0, stream>>>(
            GI + (size_t)t * BDIM * 3 * H2DIM, GH, HF,
            STATES + (size_t)t * BDIM * HDIM, H2DIM, 0);
    }

    // ---- backward encoder ----
    gemm(X, EDIM, Wih_b, bih_b, GI, 3 * H2DIM, SDIM * BDIM, 3 * H2DIM, EDIM);
    for (int t = 0; t < SDIM; ++t) {
        int s = SDIM - 1 - t;
        gemm(HB, H2DIM, Whh_b, bhh_b, GH, 3 * H2DIM, BDIM, 3 * H2DIM, H2DIM);
        gru_update<<<(BDIM * H2DIM + 255) / 256, 256, 0, stream>>>(
            GI + (size_t)s * BDIM * 3 * H2DIM, GH, HB,
            STATES + (size_t)s * BDIM * HDIM, H2DIM, H2DIM);
    }

    // ---- enc_hidden, sigmoid gate, enc_proj ----
    concat_h<<<(BDIM * HDIM + 255) / 256, 256, 0, stream>>>(HF, HB, ENCH, HDEC);
    gemm(ENCH, HDIM, W2, b2, G2, HDIM, BDIM, HDIM, HDIM);
    gemm(STATES, HDIM, W1, b1, SG, HDIM, SDIM * BDIM, HDIM, HDIM);
    {
        int total = SDIM * BDIM * HDIM;
        apply_sgate<<<(total + 255) / 256, 256, 0, stream>>>(STATES, SG, G2, total);
    }
    gemm(STATES, HDIM, We, be, EPROJ, MDIM, SDIM * BDIM, MDIM, HDIM);

    // ---- decoder ----
    for (int t = 0; t < TDIM - 1; ++t) {
        gemm(HDEC, HDIM, Wd, bd, DH, MDIM, BDIM, MDIM, HDIM);
        {
            int threads = SDIM * BDIM * 32;
            attn_scores<<<(threads + 255) / 256, 256, 0, stream>>>(DH, EPROJ, Wo, bo, SCR);
        }
        softmax_ctx<<<BDIM, 256, 0, stream>>>(SCR, STATES, CVEC);
        {
            int total = BDIM * (HDIM + EDIM);
            build_xcat<<<(total + 255) / 256, 256, 0, stream>>>(CVEC, trg, emb, XCAT, t);
        }
        gemm(XCAT, HDIM + EDIM, Wih_c, bih_c, GIC, 3 * HDIM, BDIM, 3 * HDIM, HDIM + EDIM);
        gemm(HDEC, HDIM, Whh_c, bhh_c, GHC, 3 * HDIM, BDIM, 3 * HDIM, HDIM);
        gru_update<<<(BDIM * HDIM + 255) / 256, 256, 0, stream>>>(
            GIC, GHC, HDEC, (float*)nullptr, HDIM, 0);
        gemm(HDEC, HDIM, Wv, bv, LOGITS, VDIM, BDIM, VDIM, HDIM);
        loss_row<<<BDIM, 256, 0, stream>>>(LOGITS, trg, t, NLL, WB);
        loss_accum<<<1, 64, 0, stream>>>(NLL, WB, LACC);
    }

    finalize_loss<<<1, 64, 0, stream>>>(LACC, out);
}